// HysteresisRouter_58377195487812
// MI455X (gfx1250) — compile-verified
//
#include <hip/hip_runtime.h>

// ---------------- problem constants (match reference) ----------------
#define B_DIM 16384
#define D_DIM 4096
#define E_DIM 64
#define HYST  0.5f
#define EPSC  1e-12f
// TAU == 1.0 -> logits/TAU is a no-op, folded away.

typedef float v2f __attribute__((ext_vector_type(2)));
typedef float v4f __attribute__((ext_vector_type(4)));
typedef float v8f __attribute__((ext_vector_type(8)));

// ---------------- GEMM tiling ----------------
#define KC     64      // K-chunk staged in LDS
#define MROWS  32      // rows of x per workgroup
#define PITCH  68      // LDS dword pitch: 68%64=4 -> conflict-free, 68*4%16==0 -> b128-aligned
#define XTILE  (MROWS * PITCH)   // 2176 dwords
#define WTILE  (E_DIM * PITCH)   // 4352 dwords

// async global->LDS copy of 16B per lane (CDNA5, ASYNCcnt-tracked, bypasses VGPRs)
__device__ __forceinline__ void async_copy_b128(const float* gaddr, const float* lds_ptr) {
  const unsigned int lds_off = (unsigned int)(unsigned long long)lds_ptr; // flat LDS addr[31:0]
  asm volatile("global_load_async_to_lds_b128 %0, %1, off"
               :: "v"(lds_off), "v"(gaddr) : "memory");
}
__device__ __forceinline__ void wait_async_zero() {
  asm volatile("s_wait_asynccnt 0" ::: "memory");
}

// =====================================================================
// K0: zero the three column-sum accumulators (192 floats)
// =====================================================================
__global__ __launch_bounds__(256) void init_kernel(float* __restrict__ cs) {
  const int t = threadIdx.x;
  if (t < 3 * E_DIM) cs[t] = 0.0f;
}

// =====================================================================
// K1: logits = x @ W^T + b ; probs = softmax(logits); P out; colsum0 out
// grid = B/32, block = 256 (8 waves). Wave w: rows (w&1)*16.., all 64 cols.
// Double-buffered async global->LDS staging overlapped with WMMA.
// =====================================================================
__global__ __launch_bounds__(256) void gemm_softmax_kernel(
    const float* __restrict__ x, const float* __restrict__ W,
    const float* __restrict__ bias, float* __restrict__ P,
    float* __restrict__ cs0)
{
  __shared__ float xs[2][XTILE];   // 2 x 8704 B : x tiles (buf0 reused for logits)
  __shared__ float ws[2][WTILE];   // 2 x 17408 B: W tiles
  __shared__ float colpart[E_DIM];

  const int t    = threadIdx.x;
  const int lane = t & 31;
  const int wave = t >> 5;
  const int row0 = blockIdx.x * MROWS;

  if (t < E_DIM) colpart[t] = 0.0f;

  const int mhalf = (wave & 1) * 16;     // which 16-row half this wave computes
  const int lrow  = lane & 15;           // A: M index / B: N index (within 16)
  const int khi   = (lane >> 4) * 2;     // K sub-offset per ISA A/B layout

  // issue the async loads for one K-chunk into buffer bi (6 x b128 per thread)
  auto issue_chunk = [&](int bi, int k0) {
    int f = t;
    #pragma unroll
    for (int rep = 0; rep < 2; ++rep) {        // x tile: 32 rows x 64 K
      const int r  = f >> 4;
      const int c4 = (f & 15) * 4;
      async_copy_b128(x + (size_t)(row0 + r) * D_DIM + k0 + c4,
                      &xs[bi][r * PITCH + c4]);
      f += 256;
    }
    int g = t;
    #pragma unroll
    for (int rep = 0; rep < 4; ++rep) {        // W tile: 64 rows x 64 K
      const int r  = g >> 4;
      const int c4 = (g & 15) * 4;
      async_copy_b128(W + (size_t)r * D_DIM + k0 + c4,
                      &ws[bi][r * PITCH + c4]);
      g += 256;
    }
  };

  v8f acc0 = {}, acc1 = {}, acc2 = {}, acc3 = {};

  issue_chunk(0, 0);   // prologue: fill buffer 0

  for (int k0 = 0, i = 0; k0 < D_DIM; k0 += KC, ++i) {
    const int bi = i & 1;
    wait_async_zero();     // my loads into buf[bi] complete
    __syncthreads();       // everyone's loads done; buf[bi^1] no longer being read
    if (k0 + KC < D_DIM) issue_chunk(bi ^ 1, k0 + KC);  // overlap next DMA with compute

    const float* xt = &xs[bi][0];
    const float* wt = &ws[bi][0];
    #pragma unroll 4
    for (int j = 0; j < KC; j += 4) {
      const v2f a  = *(const v2f*)(xt + (mhalf + lrow) * PITCH + j + khi);
      const v2f b0 = *(const v2f*)(wt + ( 0 + lrow) * PITCH + j + khi);
      const v2f b1 = *(const v2f*)(wt + (16 + lrow) * PITCH + j + khi);
      const v2f b2 = *(const v2f*)(wt + (32 + lrow) * PITCH + j + khi);
      const v2f b3 = *(const v2f*)(wt + (48 + lrow) * PITCH + j + khi);
      acc0 = __builtin_amdgcn_wmma_f32_16x16x4_f32(false, a, false, b0, (short)0, acc0, false, false);
      acc1 = __builtin_amdgcn_wmma_f32_16x16x4_f32(false, a, false, b1, (short)0, acc1, false, false);
      acc2 = __builtin_amdgcn_wmma_f32_16x16x4_f32(false, a, false, b2, (short)0, acc2, false, false);
      acc3 = __builtin_amdgcn_wmma_f32_16x16x4_f32(false, a, false, b3, (short)0, acc3, false, false);
    }
  }

  // ---- spill logits (+bias) into xs[0] per the C/D VGPR layout ----
  // (all waves are past the last barrier, so buf0 is no longer a WMMA source
  //  for anyone still computing the final chunk, which lives in buf1)
  {
    const int crow = mhalf + (lane >> 4) * 8;   // M = vgpr + (lane>=16 ? 8 : 0)
    const int ccol = lane & 15;                  // N = lane % 16
    const float b0 = bias[ccol +  0];
    const float b1 = bias[ccol + 16];
    const float b2 = bias[ccol + 32];
    const float b3 = bias[ccol + 48];
    #pragma unroll
    for (int i = 0; i < 8; ++i) {
      xs[0][(crow + i) * PITCH + ccol +  0] = acc0[i] + b0;
      xs[0][(crow + i) * PITCH + ccol + 16] = acc1[i] + b1;
      xs[0][(crow + i) * PITCH + ccol + 32] = acc2[i] + b2;
      xs[0][(crow + i) * PITCH + ccol + 48] = acc3[i] + b3;
    }
  }
  __syncthreads();

  // ---- row softmax: 8 threads per row, shfl_xor reductions (width 8) ----
  {
    const int r   = t >> 3;     // 0..31
    const int sub = t & 7;      // 8 cols each
    const float* lr = &xs[0][r * PITCH + sub * 8];
    float v[8];
    float m = -3.4e38f;
    #pragma unroll
    for (int j = 0; j < 8; ++j) { v[j] = lr[j]; m = fmaxf(m, v[j]); }
    #pragma unroll
    for (int off = 1; off < 8; off <<= 1) m = fmaxf(m, __shfl_xor(m, off, 8));
    float s = 0.0f;
    #pragma unroll
    for (int j = 0; j < 8; ++j) { v[j] = __expf(v[j] - m); s += v[j]; }
    #pragma unroll
    for (int off = 1; off < 8; off <<= 1) s += __shfl_xor(s, off, 8);
    const float inv = 1.0f / s;
    float* pr = P + (size_t)(row0 + r) * E_DIM + sub * 8;
    #pragma unroll
    for (int j = 0; j < 8; ++j) {
      const float p = v[j] * inv;
      pr[j] = p;
      atomicAdd(&colpart[sub * 8 + j], p);   // ds_add_f32
    }
  }
  __syncthreads();
  if (t < E_DIM) atomicAdd(&cs0[t], colpart[t]);  // global_atomic_add_f32
}

// =====================================================================
// K2/K3: one Sinkhorn iteration's row pass (uses previous colsum),
// accumulates next colsum. One thread per row; P is L2-resident.
// =====================================================================
__global__ __launch_bounds__(256) void sinkhorn_rowpass(
    float* __restrict__ P, const float* __restrict__ cs_in,
    float* __restrict__ cs_out)
{
  __shared__ float colpart[E_DIM];
  __shared__ float csl[E_DIM];
  const int t = threadIdx.x;
  if (t < E_DIM) {
    colpart[t] = 0.0f;
    csl[t] = ((float)E_DIM / (float)B_DIM) / fmaxf(cs_in[t], EPSC); // colnorm * E/B folded
  }
  __syncthreads();

  const size_t row = (size_t)blockIdx.x * blockDim.x + t;
  float* pr = P + row * E_DIM;

  float vals[E_DIM];
  float s = 0.0f;
  #pragma unroll
  for (int c4 = 0; c4 < E_DIM; c4 += 4) {
    const v4f v = *(const v4f*)(pr + c4);
    #pragma unroll
    for (int j = 0; j < 4; ++j) { const float w = v[j] * csl[c4 + j]; vals[c4 + j] = w; s += w; }
  }
  const float inv = 1.0f / fmaxf(s, EPSC);
  #pragma unroll
  for (int c4 = 0; c4 < E_DIM; c4 += 4) {
    v4f v;
    #pragma unroll
    for (int j = 0; j < 4; ++j) {
      const float w = vals[c4 + j] * inv;
      v[j] = w;
      atomicAdd(&colpart[c4 + j], w);
    }
    *(v4f*)(pr + c4) = v;
  }
  __syncthreads();
  if (t < E_DIM) atomicAdd(&cs_out[t], colpart[t]);
}

// =====================================================================
// K4: Sinkhorn iter 3 row pass + hysteresis + inhibit + renorms + top-k
// =====================================================================
__global__ __launch_bounds__(256) void finalize_kernel(
    const float* __restrict__ P, const float* __restrict__ cs_in,
    const unsigned char* __restrict__ prev_mask, const float* __restrict__ inhibit,
    const int* __restrict__ kptr,
    float* __restrict__ out_probs, float* __restrict__ out_mask)
{
  __shared__ float csl[E_DIM];
  __shared__ float inh[E_DIM];
  const int t = threadIdx.x;
  if (t < E_DIM) {
    csl[t] = ((float)E_DIM / (float)B_DIM) / fmaxf(cs_in[t], EPSC);
    inh[t] = inhibit[t];
  }
  __syncthreads();

  const size_t row = (size_t)blockIdx.x * blockDim.x + t;
  const float* pr = P + row * E_DIM;

  float vals[E_DIM];
  // Sinkhorn iteration 3 (col-normalize via csl, then row-normalize)
  float s = 0.0f;
  #pragma unroll
  for (int c4 = 0; c4 < E_DIM; c4 += 4) {
    const v4f v = *(const v4f*)(pr + c4);
    #pragma unroll
    for (int j = 0; j < 4; ++j) { const float w = v[j] * csl[c4 + j]; vals[c4 + j] = w; s += w; }
  }
  float inv = 1.0f / fmaxf(s, EPSC);

  // hysteresis blend with previous mask (bools packed: 1 byte each -> u64 words)
  const unsigned long long* pm64 = (const unsigned long long*)(prev_mask + row * E_DIM);
  unsigned long long pmw[8];
  int pcnt = 0;
  #pragma unroll
  for (int q = 0; q < 8; ++q) { pmw[q] = pm64[q]; pcnt += __popcll(pmw[q]); }
  const float pscale = HYST / fmaxf((float)pcnt, 1.0f);
  float s2 = 0.0f;
  #pragma unroll
  for (int c = 0; c < E_DIM; ++c) {
    const float m = (float)((pmw[c >> 3] >> ((c & 7) * 8)) & 1ull);
    const float w = (1.0f - HYST) * (vals[c] * inv) + m * pscale;
    vals[c] = w; s2 += w;
  }
  inv = 1.0f / fmaxf(s2, EPSC);

  // inhibition + renorm
  float s3 = 0.0f;
  #pragma unroll
  for (int c = 0; c < E_DIM; ++c) { const float w = vals[c] * inv * inh[c]; vals[c] = w; s3 += w; }
  inv = 1.0f / fmaxf(s3, EPSC);

  float* op = out_probs + row * E_DIM;
  #pragma unroll
  for (int c4 = 0; c4 < E_DIM; c4 += 4) {
    v4f v;
    #pragma unroll
    for (int j = 0; j < 4; ++j) { vals[c4 + j] *= inv; v[j] = vals[c4 + j]; }
    *(v4f*)(op + c4) = v;
  }

  // top-k selection (k is a runtime input; reference uses k=2)
  int kk = *kptr;
  kk = kk < 0 ? 0 : (kk > E_DIM ? E_DIM : kk);
  unsigned long long sel = 0ull;
  for (int i = 0; i < kk; ++i) {
    float best = -3.4e38f; int bi = 0;
    #pragma unroll
    for (int c = 0; c < E_DIM; ++c) {
      const bool avail = ((sel >> c) & 1ull) == 0ull;
      if (avail && vals[c] > best) { best = vals[c]; bi = c; }
    }
    sel |= 1ull << bi;
  }
  float* om = out_mask + row * E_DIM;
  #pragma unroll
  for (int c = 0; c < E_DIM; ++c) om[c] = (float)((sel >> c) & 1ull);
}

// =====================================================================
extern "C" void kernel_launch(void* const* d_in, const int* in_sizes, int n_in,
                              void* d_out, int out_size, void* d_ws, size_t ws_size,
                              hipStream_t stream) {
  (void)in_sizes; (void)n_in; (void)out_size; (void)ws_size;
  const float*         x         = (const float*)d_in[0];
  const float*         W         = (const float*)d_in[1];
  const float*         b         = (const float*)d_in[2];
  const unsigned char* prev_mask = (const unsigned char*)d_in[3]; // bool array
  const float*         inhibit   = (const float*)d_in[4];
  const int*           kptr      = (const int*)d_in[5];

  float* P  = (float*)d_ws;                       // 16384*64 f32 = 4 MB scratch
  float* cs = P + (size_t)B_DIM * E_DIM;          // 3 * 64 colsum accumulators

  float* out_probs = (float*)d_out;
  float* out_mask  = out_probs + (size_t)B_DIM * E_DIM;

  init_kernel<<<1, 256, 0, stream>>>(cs);
  gemm_softmax_kernel<<<B_DIM / MROWS, 256, 0, stream>>>(x, W, b, P, cs);
  sinkhorn_rowpass<<<B_DIM / 256, 256, 0, stream>>>(P, cs,       cs + 64);
  sinkhorn_rowpass<<<B_DIM / 256, 256, 0, stream>>>(P, cs + 64,  cs + 128);
  finalize_kernel<<<B_DIM / 256, 256, 0, stream>>>(P, cs + 128, prev_mask, inhibit,
                                                   kptr, out_probs, out_mask);
}